// GNN_24326694765162
// MI455X (gfx1250) — compile-verified
//
#include <hip/hip_runtime.h>

typedef float v2f __attribute__((ext_vector_type(2)));
typedef float v8f __attribute__((ext_vector_type(8)));

#define IN_C   17
#define HID_C  64
#define OUT_C  32
#define NGRAPH 512

// ---------------- utility kernels ----------------

__global__ void k_zero4(float4* __restrict__ p, long n4) {
    long i = (long)blockIdx.x * blockDim.x + threadIdx.x;
    if (i < n4) p[i] = make_float4(0.f, 0.f, 0.f, 0.f);
}

__global__ void k_init_deg(float* __restrict__ deg, int n) {
    int i = blockIdx.x * blockDim.x + threadIdx.x;
    if (i < n) deg[i] = 1.0f;                     // self-loop contributes 1
}

__global__ void k_deg_count(const int* __restrict__ dst, float* __restrict__ deg, int E) {
    int e = blockIdx.x * blockDim.x + threadIdx.x;
    if (e < E) atomicAdd(&deg[dst[e]], 1.0f);
}

__global__ void k_dinv(const float* __restrict__ deg, float* __restrict__ dinv, int n) {
    int i = blockIdx.x * blockDim.x + threadIdx.x;
    if (i < n) dinv[i] = rsqrtf(deg[i]);          // deg >= 1 always
}

// ---------------- WMMA GEMM: out[N,COUT] = A[N,CIN] @ W[CIN,COUT] ----------------
// one wave (32 threads) per 16-row tile; f32 16x16x4 WMMA

template <int CIN, int COUT>
__global__ void k_gemm(const float* __restrict__ A, const float* __restrict__ W,
                       float* __restrict__ out, int n) {
    const int  lane = threadIdx.x;
    const int  m    = lane & 15;
    const int  h    = lane >> 4;
    const long rowbase = (long)blockIdx.x * 16;

    constexpr int NT = COUT / 16;
    constexpr int KS = (CIN + 3) / 4;

    // clamped row for loads (duplicate row, masked out at store)
    const long rA = rowbase + m < n ? rowbase + m : (long)n - 1;
    const float* ar = A + rA * CIN;

    v8f acc[NT] = {};
    for (int ks = 0; ks < KS; ++ks) {
        const int ka = ks * 4 + 2 * h;
        const int kb = ka + 1;
        const float ma = (ka < CIN) ? 1.f : 0.f;
        const float mb = (kb < CIN) ? 1.f : 0.f;
        const int kca = (ka < CIN) ? ka : (CIN - 1);
        const int kcb = (kb < CIN) ? kb : (CIN - 1);

        v2f a;
        a.x = ar[kca] * ma;
        a.y = ar[kcb] * mb;

#pragma unroll
        for (int nt = 0; nt < NT; ++nt) {
            v2f b;
            b.x = W[kca * COUT + nt * 16 + m] * ma;
            b.y = W[kcb * COUT + nt * 16 + m] * mb;
            acc[nt] = __builtin_amdgcn_wmma_f32_16x16x4_f32(
                false, a, false, b, (short)0, acc[nt], false, false);
        }
    }
#pragma unroll
    for (int nt = 0; nt < NT; ++nt)
#pragma unroll
        for (int v = 0; v < 8; ++v) {
            long r = rowbase + v + 8 * h;
            if (r < n) out[r * COUT + nt * 16 + m] = acc[nt][v];
        }
}

// ---------------- edge aggregation: agg[dst] += dinv[s]*dinv[d] * h[src] ----------------
// thread = (edge, 4-channel group); 16 (or 8) lanes share one edge -> coalesced row gather

template <int C>
__global__ void k_edge_agg(const int* __restrict__ src, const int* __restrict__ dst,
                           const float* __restrict__ dinv, const float* __restrict__ hin,
                           float* __restrict__ agg, int E) {
    constexpr int G = C / 4;
    long tid = (long)blockIdx.x * blockDim.x + threadIdx.x;
    if (tid >= (long)E * G) return;
    int e  = (int)(tid / G);
    int cg = (int)(tid % G);
    int s = src[e], d = dst[e];
    float w = dinv[s] * dinv[d];
    const float4 v = *(const float4*)(hin + (long)s * C + cg * 4);
    float* o = agg + (long)d * C + cg * 4;
    atomicAdd(o + 0, w * v.x);
    atomicAdd(o + 1, w * v.y);
    atomicAdd(o + 2, w * v.z);
    atomicAdd(o + 3, w * v.w);
}

// ---------------- layer-1 epilogue: add self-loop + bias, ReLU (in place) ----------------

__global__ void k_post1(float* __restrict__ agg1, const float* __restrict__ h1,
                        const float* __restrict__ dinv, const float* __restrict__ b1, int n) {
    long tid = (long)blockIdx.x * blockDim.x + threadIdx.x;
    if (tid >= (long)n * HID_C) return;
    int i = (int)(tid / HID_C);
    int c = (int)(tid % HID_C);
    float sl = dinv[i];
    float v = agg1[tid] + sl * sl * h1[tid] + b1[c];
    agg1[tid] = v > 0.f ? v : 0.f;
}

// ---------------- pooling: pool[g] += agg2[i] + dinv[i]^2 * h2[i]  ----------------

__global__ void k_cnt(const int* __restrict__ batch, float* __restrict__ cnt, int n) {
    int i = blockIdx.x * blockDim.x + threadIdx.x;
    if (i < n) atomicAdd(&cnt[batch[i]], 1.0f);
}

__global__ void k_pool(const float* __restrict__ agg2, const float* __restrict__ h2,
                       const float* __restrict__ dinv, const int* __restrict__ batch,
                       float* __restrict__ pool, int n) {
    long tid = (long)blockIdx.x * blockDim.x + threadIdx.x;
    if (tid >= (long)n * OUT_C) return;
    int i = (int)(tid / OUT_C);
    int c = (int)(tid % OUT_C);
    float sl = dinv[i];
    float v = agg2[tid] + sl * sl * h2[tid];
    atomicAdd(&pool[batch[i] * OUT_C + c], v);
}

// ---------------- FC head: out = relu(pooled@Wf1+bf1)@Wf2+bf2, pooled = pool/cnt + b2 ----
// one wave per 16-graph tile; LDS used to convert C-frag layout -> A-frag layout

__global__ void k_fc(const float* __restrict__ pool, const float* __restrict__ cnt,
                     const float* __restrict__ b2,
                     const float* __restrict__ Wf1, const float* __restrict__ bf1,
                     const float* __restrict__ Wf2, const float* __restrict__ bf2,
                     float* __restrict__ out) {
    __shared__ float P[16 * 32];
    __shared__ float T[16 * 32];
    const int lane = threadIdx.x;
    const int m = lane & 15;
    const int h = lane >> 4;
    const int g0 = blockIdx.x * 16;

    // stage pooled means (+b2, commuted through the mean) into LDS
    for (int idx = lane; idx < 16 * 32; idx += 32) {
        int r = idx >> 5, c = idx & 31;
        float ct = cnt[g0 + r];
        ct = ct > 1.f ? ct : 1.f;
        P[idx] = pool[(g0 + r) * 32 + c] / ct + b2[c];
    }
    __syncthreads();

    v8f acc[2] = {};
#pragma unroll
    for (int ks = 0; ks < 8; ++ks) {
        const int ka = ks * 4 + 2 * h, kb = ka + 1;
        v2f a; a.x = P[m * 32 + ka]; a.y = P[m * 32 + kb];
#pragma unroll
        for (int nt = 0; nt < 2; ++nt) {
            v2f b; b.x = Wf1[ka * 32 + nt * 16 + m]; b.y = Wf1[kb * 32 + nt * 16 + m];
            acc[nt] = __builtin_amdgcn_wmma_f32_16x16x4_f32(
                false, a, false, b, (short)0, acc[nt], false, false);
        }
    }
#pragma unroll
    for (int nt = 0; nt < 2; ++nt)
#pragma unroll
        for (int v = 0; v < 8; ++v) {
            int r = v + 8 * h, c = nt * 16 + m;
            float t = acc[nt][v] + bf1[c];
            T[r * 32 + c] = t > 0.f ? t : 0.f;
        }
    __syncthreads();

    v8f o[2] = {};
#pragma unroll
    for (int ks = 0; ks < 8; ++ks) {
        const int ka = ks * 4 + 2 * h, kb = ka + 1;
        v2f a; a.x = T[m * 32 + ka]; a.y = T[m * 32 + kb];
#pragma unroll
        for (int nt = 0; nt < 2; ++nt) {
            v2f b; b.x = Wf2[ka * 32 + nt * 16 + m]; b.y = Wf2[kb * 32 + nt * 16 + m];
            o[nt] = __builtin_amdgcn_wmma_f32_16x16x4_f32(
                false, a, false, b, (short)0, o[nt], false, false);
        }
    }
#pragma unroll
    for (int nt = 0; nt < 2; ++nt)
#pragma unroll
        for (int v = 0; v < 8; ++v) {
            int r = v + 8 * h, c = nt * 16 + m;
            out[(g0 + r) * 32 + c] = o[nt][v] + bf2[c];
        }
}

// ---------------- launcher ----------------

extern "C" void kernel_launch(void* const* d_in, const int* in_sizes, int n_in,
                              void* d_out, int out_size, void* d_ws, size_t ws_size,
                              hipStream_t stream) {
    const float* x     = (const float*)d_in[0];
    const int*   ei    = (const int*)d_in[1];
    const int*   batch = (const int*)d_in[2];
    const float* W1  = (const float*)d_in[3];
    const float* b1  = (const float*)d_in[4];
    const float* W2  = (const float*)d_in[5];
    const float* b2  = (const float*)d_in[6];
    const float* Wf1 = (const float*)d_in[7];
    const float* bf1 = (const float*)d_in[8];
    const float* Wf2 = (const float*)d_in[9];
    const float* bf2 = (const float*)d_in[10];

    const int N = in_sizes[0] / IN_C;
    const int E = in_sizes[1] / 2;
    const int* src = ei;
    const int* dst = ei + E;

    float* ws   = (float*)d_ws;
    float* deg  = ws;                              // N
    float* dinv = deg  + N;                        // N
    float* h1   = dinv + N;                        // N*64
    float* agg1 = h1   + (size_t)N * HID_C;        // N*64
    float* h2   = agg1 + (size_t)N * HID_C;        // N*32
    float* agg2 = h2   + (size_t)N * OUT_C;        // N*32
    float* pool = agg2 + (size_t)N * OUT_C;        // 512*32
    float* cnt  = pool + NGRAPH * OUT_C;           // 512

    const int B = 256;
    // zero the atomic accumulators (agg1, agg2 adjacent? no — zero separately) + pool/cnt
    {
        long n4 = (long)N * HID_C / 4;
        k_zero4<<<(int)((n4 + B - 1) / B), B, 0, stream>>>((float4*)agg1, n4);
        n4 = (long)N * OUT_C / 4;
        k_zero4<<<(int)((n4 + B - 1) / B), B, 0, stream>>>((float4*)agg2, n4);
        n4 = (NGRAPH * OUT_C + NGRAPH) / 4;
        k_zero4<<<(int)((n4 + B - 1) / B), B, 0, stream>>>((float4*)pool, n4);
    }

    k_init_deg <<<(N + B - 1) / B, B, 0, stream>>>(deg, N);
    k_deg_count<<<(E + B - 1) / B, B, 0, stream>>>(dst, deg, E);
    k_dinv     <<<(N + B - 1) / B, B, 0, stream>>>(deg, dinv, N);

    const int tiles = (N + 15) / 16;
    k_gemm<IN_C, HID_C><<<tiles, 32, 0, stream>>>(x, W1, h1, N);

    long t1 = (long)E * (HID_C / 4);
    k_edge_agg<HID_C><<<(int)((t1 + B - 1) / B), B, 0, stream>>>(src, dst, dinv, h1, agg1, E);

    long t2 = (long)N * HID_C;
    k_post1<<<(int)((t2 + B - 1) / B), B, 0, stream>>>(agg1, h1, dinv, b1, N);

    k_gemm<HID_C, OUT_C><<<tiles, 32, 0, stream>>>(agg1, W2, h2, N);

    long t3 = (long)E * (OUT_C / 4);
    k_edge_agg<OUT_C><<<(int)((t3 + B - 1) / B), B, 0, stream>>>(src, dst, dinv, h2, agg2, E);

    k_cnt<<<(N + B - 1) / B, B, 0, stream>>>(batch, cnt, N);

    long t4 = (long)N * OUT_C;
    k_pool<<<(int)((t4 + B - 1) / B), B, 0, stream>>>(agg2, h2, dinv, batch, pool, N);

    k_fc<<<NGRAPH / 16, 32, 0, stream>>>(pool, cnt, b2, Wf1, bf1, Wf2, bf2, (float*)d_out);
}